// NeuralODEfunc_30511447670826
// MI455X (gfx1250) — compile-verified
//
#include <hip/hip_runtime.h>
#include <math.h>

// ---------------------------------------------------------------------------
// Geodesic RHS for Brill-Lindquist conformally flat metric (2 punctures).
//
// Reference math reduces analytically to (per point, spatial a in {1,2,3}):
//   p      = psi^4 at x,          psi = 1 + m0/(2 r0) + m1/(2 r1)
//   d_a    = central FD of p along axis a with EPS=1e-4  (matches reference)
//   dv_a   = -(1/(2p)) * ( 2 v_a (d.v) - d_a |v|^2 ),  dv_0 = 0
//   dx     = v
// Output layout: out[0 : 4B] = dx, out[4B : 8B] = dv   (stack([dx, dv]))
// ---------------------------------------------------------------------------

struct BLParams {
    float m0, m1;
    float p0x, p0y, p0z;
    float p1x, p1y, p1z;
};

__device__ __forceinline__ float psi4_at(float sx, float sy, float sz,
                                         const BLParams& P) {
    // puncture 0
    float ax = sx - P.p0x, ay = sy - P.p0y, az = sz - P.p0z;
    float r0 = sqrtf(ax * ax + ay * ay + az * az);
    r0 += (r0 == 0.0f) ? 1e-8f : 0.0f;   // reference's r==0 guard
    // puncture 1
    float bx = sx - P.p1x, by = sy - P.p1y, bz = sz - P.p1z;
    float r1 = sqrtf(bx * bx + by * by + bz * bz);
    r1 += (r1 == 0.0f) ? 1e-8f : 0.0f;
    float psi = 1.0f + (__fdividef(P.m0, 2.0f * r0) +
                        __fdividef(P.m1, 2.0f * r1));
    float s = psi * psi;
    return s * s;
}

// One-thread prelude: softplus the masses (jax-stable form) and pack the
// 8 uniform parameters contiguously so the hot kernel does two b128 loads.
__global__ void bl_prep_kernel(const float* __restrict__ pos,   // (2,3)
                               const float* __restrict__ mps,   // (2,)
                               float* __restrict__ ws) {
    if (blockIdx.x == 0 && threadIdx.x == 0) {
        float a = mps[0], b = mps[1];
        // jax.nn.softplus: log1p(exp(-|x|)) + max(x, 0)
        ws[0] = log1pf(expf(-fabsf(a))) + fmaxf(a, 0.0f);
        ws[1] = log1pf(expf(-fabsf(b))) + fmaxf(b, 0.0f);
        ws[2] = pos[0]; ws[3] = pos[1]; ws[4] = pos[2];   // puncture 0
        ws[5] = pos[3]; ws[6] = pos[4]; ws[7] = pos[5];   // puncture 1
    }
}

__global__ __launch_bounds__(256) void bl_geodesic_kernel(
    const float* __restrict__ xs,   // x_state (2, B, 4) flat
    const float* __restrict__ ws,   // [m0, m1, p0x, p0y, p0z, p1x, p1y, p1z]
    float* __restrict__ out,        // (2, B, 4) flat
    int B) {
    const int b = blockIdx.x * blockDim.x + threadIdx.x;

    // Uniform parameters: two 128-bit loads, broadcast via cache.
    const float4 pa = ((const float4*)ws)[0];
    const float4 pb = ((const float4*)ws)[1];
    BLParams P;
    P.m0 = pa.x;  P.m1 = pa.y;
    P.p0x = pa.z; P.p0y = pa.w; P.p0z = pb.x;
    P.p1x = pb.y; P.p1y = pb.z; P.p1z = pb.w;

    if (b >= B) return;

    const float4 X = ((const float4*)xs)[b];                 // (t, x, y, z)
    const float4 V = ((const float4*)xs)[(size_t)B + b];     // v

    const float x = X.y, y = X.z, z = X.w;

    const float EPSf  = 1e-4f;
    const float INV2E = 1.0f / (2.0f * 1e-4f);

    // psi^4 at center + 6 offset points (same FD scheme as reference;
    // the time-axis FD is identically zero and is skipped).
    const float p   = psi4_at(x, y, z, P);
    const float dpx = (psi4_at(x + EPSf, y, z, P) -
                       psi4_at(x - EPSf, y, z, P)) * INV2E;
    const float dpy = (psi4_at(x, y + EPSf, z, P) -
                       psi4_at(x, y - EPSf, z, P)) * INV2E;
    const float dpz = (psi4_at(x, y, z + EPSf, P) -
                       psi4_at(x, y, z - EPSf, P)) * INV2E;

    const float vx = V.y, vy = V.z, vz = V.w;
    const float dvdot = dpx * vx + dpy * vy + dpz * vz;
    const float vsq   = vx * vx + vy * vy + vz * vz;
    const float cinv  = -0.5f * __fdividef(1.0f, p);

    float4 DV;
    DV.x = 0.0f;                                  // dv_t == 0 exactly
    DV.y = cinv * (2.0f * vx * dvdot - dpx * vsq);
    DV.z = cinv * (2.0f * vy * dvdot - dpy * vsq);
    DV.w = cinv * (2.0f * vz * dvdot - dpz * vsq);

    ((float4*)out)[b]             = V;            // dx = v
    ((float4*)out)[(size_t)B + b] = DV;           // dv
}

extern "C" void kernel_launch(void* const* d_in, const int* in_sizes, int n_in,
                              void* d_out, int out_size, void* d_ws,
                              size_t ws_size, hipStream_t stream) {
    // d_in[0] = t (unused scalar)
    const float* xs  = (const float*)d_in[1];   // x_state (2, B, 4)
    const float* pos = (const float*)d_in[2];   // positions (2, 3)
    const float* mps = (const float*)d_in[3];   // masses_presoftplus (2,)
    float* out = (float*)d_out;
    float* ws  = (float*)d_ws;

    const int B = in_sizes[1] / 8;              // (2, B, 4) -> B

    bl_prep_kernel<<<1, 1, 0, stream>>>(pos, mps, ws);

    const int threads = 256;                    // 8 wave32 per workgroup
    const int blocks  = (B + threads - 1) / threads;
    bl_geodesic_kernel<<<blocks, threads, 0, stream>>>(xs, ws, out, B);
}